// HybridQLSTMQuantum_65481071409905
// MI455X (gfx1250) — compile-verified
//
#include <hip/hip_runtime.h>

#define S_LEN  512
#define B_SZ   1024
#define VOCAB  32000
#define E_DIM  64
#define LATENT 32
#define HID    4
#define NQ     4
#define TAG    128

typedef __attribute__((ext_vector_type(2))) float v2f;
typedef __attribute__((ext_vector_type(8))) float v8f;

__device__ __forceinline__ float fast_sigmoid(float x) {
  return 1.0f / (1.0f + __expf(-x));
}
__device__ __forceinline__ float fast_tanh(float x) {
  // args are bounded (|x| < ~4) in this model, exp form is safe
  float t = __expf(2.0f * x);
  return (t - 1.0f) / (t + 1.0f);
}

// ---------------------------------------------------------------------------
// Kernel A: per-token count of embedding values > 0.5 (feat depends only on
// the token id, so the whole "quanv filter" collapses to this table).
// ---------------------------------------------------------------------------
__global__ void tok_count_kernel(const float* __restrict__ emb,
                                 int* __restrict__ count) {
  int t = blockIdx.x * blockDim.x + threadIdx.x;
  if (t >= VOCAB) return;
  const float4* row = reinterpret_cast<const float4*>(emb + (size_t)t * E_DIM);
  int c = 0;
#pragma unroll
  for (int i = 0; i < E_DIM / 4; ++i) {
    float4 v = row[i];
    c += (v.x > 0.5f) + (v.y > 0.5f) + (v.z > 0.5f) + (v.w > 0.5f);
  }
  count[t] = c;
}

// ---------------------------------------------------------------------------
// Kernel B: feat has only 65 possible values (k/64). Fold the whole AE MLP and
// the z-part of the 4 gate linears (+bias+theta) into a 65 x 16 table.
// pre[k][g*4+q] = (AE(k/64) @ Wg[:32]) + bg + thg
// ---------------------------------------------------------------------------
__global__ void ae_table_kernel(
    const float* __restrict__ w1, const float* __restrict__ b1,
    const float* __restrict__ w2, const float* __restrict__ b2,
    const float* __restrict__ w3, const float* __restrict__ b3,
    const float* __restrict__ wf, const float* __restrict__ bf,
    const float* __restrict__ wi, const float* __restrict__ bi,
    const float* __restrict__ wg, const float* __restrict__ bg,
    const float* __restrict__ wo, const float* __restrict__ bo,
    const float* __restrict__ thf, const float* __restrict__ thi,
    const float* __restrict__ thg, const float* __restrict__ tho,
    float* __restrict__ pre) {
  int k = threadIdx.x;
  if (k > 64) return;
  float f = (float)k * (1.0f / 64.0f);
  float h1[64];
  for (int j = 0; j < 64; ++j) h1[j] = fmaxf(f * w1[j] + b1[j], 0.0f);
  float h2[32];
  for (int m = 0; m < 32; ++m) {
    float s = b2[m];
    for (int j = 0; j < 64; ++j) s += h1[j] * w2[j * 32 + m];
    h2[m] = fmaxf(s, 0.0f);
  }
  float z[32];
  for (int l = 0; l < 32; ++l) {
    float s = b3[l];
    for (int m = 0; m < 32; ++m) s += h2[m] * w3[m * 32 + l];
    z[l] = s;
  }
  const float* W[4]  = {wf, wi, wg, wo};
  const float* Bv[4] = {bf, bi, bg, bo};
  const float* Th[4] = {thf, thi, thg, tho};
  for (int g = 0; g < 4; ++g)
    for (int q = 0; q < NQ; ++q) {
      float s = Bv[g][q] + Th[g][q];
      for (int l = 0; l < LATENT; ++l) s += z[l] * W[g][l * NQ + q];
      pre[k * 16 + g * 4 + q] = s;
    }
}

// ---------------------------------------------------------------------------
// Kernel C: sequential QLSTM scan. One thread per batch element (1024 lanes
// in parallel). Per step: table lookup by token count, 4x4 hidden GEMV per
// gate, closed-form qgate (products of native cosines), bounded sigmoid/tanh.
// ---------------------------------------------------------------------------
__global__ void __launch_bounds__(256) qlstm_scan_kernel(
    const int* __restrict__ sentence, const int* __restrict__ count,
    const float* __restrict__ pre,
    const float* __restrict__ wf, const float* __restrict__ wi,
    const float* __restrict__ wg, const float* __restrict__ wo,
    float* __restrict__ outs) {
  __shared__ float sPre[65 * 16];
  for (int i = threadIdx.x; i < 65 * 16; i += blockDim.x) sPre[i] = pre[i];
  __syncthreads();

  // hidden-part weight rows 32..35 (uniform across lanes -> scalar loads)
  float wh[4][4][4];
  const float* Wp[4] = {wf, wi, wg, wo};
#pragma unroll
  for (int g = 0; g < 4; ++g)
#pragma unroll
    for (int j = 0; j < 4; ++j)
#pragma unroll
      for (int q = 0; q < 4; ++q)
        wh[g][j][q] = Wp[g][(LATENT + j) * NQ + q];

  int b = blockIdx.x * blockDim.x + threadIdx.x;  // 0..1023
  float h[4] = {0.f, 0.f, 0.f, 0.f};
  float cst[4] = {0.f, 0.f, 0.f, 0.f};

  int k = count[sentence[b]];  // software-pipelined token-count lookup
  for (int s = 0; s < S_LEN; ++s) {
    int knext = 0;
    if (s + 1 < S_LEN) knext = count[sentence[(size_t)(s + 1) * B_SZ + b]];

    const float* p = &sPre[k * 16];
    float qz[4][4];
#pragma unroll
    for (int g = 0; g < 4; ++g) {
      float a[4];
#pragma unroll
      for (int q = 0; q < 4; ++q)
        a[q] = p[g * 4 + q] + h[0] * wh[g][0][q] + h[1] * wh[g][1][q] +
               h[2] * wh[g][2][q] + h[3] * wh[g][3][q];
      float c0 = __cosf(a[0]), c1 = __cosf(a[1]);
      float c2 = __cosf(a[2]), c3 = __cosf(a[3]);
      float z1 = c0 * c1, z2 = z1 * c2, z3 = z2 * c3, z0 = c1 * c2 * c3;
      qz[g][0] = z0; qz[g][1] = z1; qz[g][2] = z2; qz[g][3] = z3;
    }
    float hn[4];
#pragma unroll
    for (int q = 0; q < 4; ++q) {
      float fg = fast_sigmoid(qz[0][q]);
      float ig = fast_sigmoid(qz[1][q]);
      float gg = fast_tanh(qz[2][q]);
      float og = fast_sigmoid(qz[3][q]);
      float cn = fg * cst[q] + ig * gg;
      cst[q] = cn;
      hn[q] = og * fast_tanh(cn);
    }
    h[0] = hn[0]; h[1] = hn[1]; h[2] = hn[2]; h[3] = hn[3];
    float4 hv;  hv.x = hn[0]; hv.y = hn[1]; hv.z = hn[2]; hv.w = hn[3];
    reinterpret_cast<float4*>(outs)[(size_t)s * B_SZ + b] = hv;  // b128 store
    k = knext;
  }
}

// ---------------------------------------------------------------------------
// Kernel D: fused  logits = outs @ w_tag + b_tag  ->  log_softmax.
// One wave per 16-row tile: 8 x V_WMMA_F32_16X16X4_F32 (TAG=128 = 8 N-tiles),
// then in-register row reductions via width-16 xor shuffles (C/D layout keeps
// each row inside one half-wave), then a single streamed 256 MB output pass.
// ---------------------------------------------------------------------------
__global__ void __launch_bounds__(256) tag_logsoftmax_kernel(
    const float* __restrict__ outs, const float* __restrict__ w_tag,
    const float* __restrict__ b_tag, float* __restrict__ out) {
  const int lane = threadIdx.x & 31;
  const int wave = threadIdx.x >> 5;
  const int tileRow = blockIdx.x * 8 + wave;  // 16-row tile index
  const int rowBase = tileRow * 16;
  const int n = lane & 15;                 // N (or M) selector inside tile
  const int khalf = (lane >> 4) * 2;       // lanes 0-15: K0/K1, lanes 16-31: K2/K3

  // A fragment: 16x4 rows of outs; one global_load_b64 per lane
  v2f a = *reinterpret_cast<const v2f*>(outs + (size_t)(rowBase + n) * HID + khalf);

  v8f d[8];
#pragma unroll
  for (int t = 0; t < 8; ++t) {
    v2f bm;
    bm.x = w_tag[(khalf)*TAG + t * 16 + n];
    bm.y = w_tag[(khalf + 1) * TAG + t * 16 + n];
    float bt = b_tag[t * 16 + n];
    v8f c = {bt, bt, bt, bt, bt, bt, bt, bt};
    d[t] = __builtin_amdgcn_wmma_f32_16x16x4_f32(
        /*neg_a=*/false, a, /*neg_b=*/false, bm,
        /*c_mod=*/(short)0, c, /*reuse_a=*/false, /*reuse_b=*/false);
  }

  // log-softmax over the 128 tags of each row.
  // Row for (lane, vgpr v): M = v + (lane>=16 ? 8 : 0); column = t*16 + n.
#pragma unroll
  for (int v = 0; v < 8; ++v) {
    float m = d[0][v];
#pragma unroll
    for (int t = 1; t < 8; ++t) m = fmaxf(m, d[t][v]);
#pragma unroll
    for (int off = 1; off < 16; off <<= 1) m = fmaxf(m, __shfl_xor(m, off, 16));
    float ssum = 0.0f;
#pragma unroll
    for (int t = 0; t < 8; ++t) ssum += __expf(d[t][v] - m);
#pragma unroll
    for (int off = 1; off < 16; off <<= 1) ssum += __shfl_xor(ssum, off, 16);
    float lse = m + __logf(ssum);

    size_t row = (size_t)rowBase + v + ((lane >> 4) << 3);
    float* orow = out + row * TAG;
#pragma unroll
    for (int t = 0; t < 8; ++t) orow[t * 16 + n] = d[t][v] - lse;
  }
}

// ---------------------------------------------------------------------------
extern "C" void kernel_launch(void* const* d_in, const int* in_sizes, int n_in,
                              void* d_out, int out_size, void* d_ws,
                              size_t ws_size, hipStream_t stream) {
  (void)in_sizes; (void)n_in; (void)out_size; (void)ws_size;

  const int*   sentence = (const int*)d_in[0];
  const float* emb   = (const float*)d_in[1];
  const float* ae_w1 = (const float*)d_in[2];
  const float* ae_b1 = (const float*)d_in[3];
  const float* ae_w2 = (const float*)d_in[4];
  const float* ae_b2 = (const float*)d_in[5];
  const float* ae_w3 = (const float*)d_in[6];
  const float* ae_b3 = (const float*)d_in[7];
  const float* wf = (const float*)d_in[8];
  const float* bf = (const float*)d_in[9];
  const float* wi = (const float*)d_in[10];
  const float* bi = (const float*)d_in[11];
  const float* wg = (const float*)d_in[12];
  const float* bg = (const float*)d_in[13];
  const float* wo = (const float*)d_in[14];
  const float* bo = (const float*)d_in[15];
  const float* thf = (const float*)d_in[16];
  const float* thi = (const float*)d_in[17];
  const float* thg = (const float*)d_in[18];
  const float* tho = (const float*)d_in[19];
  const float* w_tag = (const float*)d_in[20];
  const float* b_tag = (const float*)d_in[21];
  float* out = (float*)d_out;

  char* ws = (char*)d_ws;
  int*   d_count = (int*)ws;                       // 32000 ints   (128 KB)
  float* d_pre   = (float*)(ws + 128 * 1024);      // 65*16 floats (~4 KB)
  float* d_outs  = (float*)(ws + 256 * 1024);      // 512*1024*4 f (8 MB)

  tok_count_kernel<<<(VOCAB + 255) / 256, 256, 0, stream>>>(emb, d_count);
  ae_table_kernel<<<1, 128, 0, stream>>>(ae_w1, ae_b1, ae_w2, ae_b2, ae_w3,
                                         ae_b3, wf, bf, wi, bi, wg, bg, wo, bo,
                                         thf, thi, thg, tho, d_pre);
  qlstm_scan_kernel<<<B_SZ / 256, 256, 0, stream>>>(sentence, d_count, d_pre,
                                                    wf, wi, wg, wo, d_outs);
  // (S*B/16) 16-row tiles, 8 waves (tiles) per 256-thread block
  tag_logsoftmax_kernel<<<(S_LEN * B_SZ / 16) / 8, 256, 0, stream>>>(
      d_outs, w_tag, b_tag, out);
}